// LoRALinear_21225728377224
// MI455X (gfx1250) — compile-verified
//
#include <hip/hip_runtime.h>

// LoRA linear: out[m,n] = sum_k x[m,k]*W[n,k] + bias[n] + sum_r t[m,r]*lora_A[n,r]
// where t[m,r] = sum_k x[m,k]*lora_B[r,k]   (rank-16 factorized delta)

#define IN_F   4096
#define OUT_F  4096
#define RANK   16
#define M_TOK  8192   // B*S = 4*2048

#define BM 64
#define BN 128
#define KT 32
#define LDS_STRIDE 36  // 32 + 4 pad: keeps float4 16B-aligned, conflict-free frags

typedef __attribute__((ext_vector_type(2))) float v2f;
typedef __attribute__((ext_vector_type(4))) float v4f;
typedef __attribute__((ext_vector_type(8))) float v8f;

static __device__ __forceinline__ v8f wmma_f32(v2f a, v2f b, v8f c) {
  // D(16x16,f32) = A(16x4,f32) * B(4x16,f32) + C
  return __builtin_amdgcn_wmma_f32_16x16x4_f32(false, a, false, b, (short)0, c,
                                               false, false);
}

// CDNA5 async direct global -> LDS copy, 16B per lane, tracked by ASYNCcnt.
static __device__ __forceinline__ void async_b128(float* lds_ptr,
                                                  const float* gptr) {
  // Low 32 bits of a generic LDS pointer = wave-relative LDS byte offset.
  unsigned lds_off = (unsigned)(unsigned long long)lds_ptr;
  asm volatile("global_load_async_to_lds_b128 %0, %1, off"
               :
               : "v"(lds_off), "v"(gptr)
               : "memory");
}

static __device__ __forceinline__ void wait_async_le6() {
  asm volatile("s_wait_asynccnt 0x6" ::: "memory");
}
static __device__ __forceinline__ void wait_async_0() {
  asm volatile("s_wait_asynccnt 0x0" ::: "memory");
}

// ---------------------------------------------------------------------------
// Kernel 1: t[m,r] = sum_k x[m,k] * lora_B[r,k]   t: [M_TOK, RANK]
// One wave per 16-row block of x; N = RANK = 16 is exactly one WMMA tile.
// ---------------------------------------------------------------------------
__global__ __launch_bounds__(32) void lora_t_kernel(const float* __restrict__ x,
                                                    const float* __restrict__ loraB,
                                                    float* __restrict__ t) {
  const int m0   = blockIdx.x * 16;
  const int lane = threadIdx.x & 31;
  const int half = lane >> 4;   // 0: K+0..1, 1: K+2..3 (ISA 32-bit A/B layout)
  const int l16  = lane & 15;

  const float* xrow = x     + (size_t)(m0 + l16) * IN_F;  // A row = M
  const float* brow = loraB + (size_t)l16 * IN_F;         // B col = rank r

  v8f acc = {};
  for (int k = 0; k < IN_F; k += 4) {
    const int kb = k + half * 2;
    v2f a; a.x = xrow[kb]; a.y = xrow[kb + 1];
    v2f b; b.x = brow[kb]; b.y = brow[kb + 1];
    acc = wmma_f32(a, b, acc);
  }
  // C layout: VGPR v -> row (v + half*8), col l16
#pragma unroll
  for (int v = 0; v < 8; ++v)
    t[(size_t)(m0 + half * 8 + v) * RANK + l16] = acc[v];
}

// ---------------------------------------------------------------------------
// Kernel 2: fused base GEMM + bias + rank-16 LoRA epilogue.
// Block: 256 threads = 8 waves; tile 64(M) x 128(N); wave tile 32x32 (2x2 WMMA).
// Double-buffered LDS fed by GLOBAL_LOAD_ASYNC_TO_LDS_B128 (ASYNCcnt pipeline).
// ---------------------------------------------------------------------------
__global__ __launch_bounds__(256) void lora_gemm_kernel(
    const float* __restrict__ x, const float* __restrict__ W,
    const float* __restrict__ bias, const float* __restrict__ loraA,
    const float* __restrict__ t, float* __restrict__ out) {
  __shared__ float xs[2][BM * LDS_STRIDE];   // 2 x  9216 B
  __shared__ float ws[2][BN * LDS_STRIDE];   // 2 x 18432 B

  const int tid  = threadIdx.x;
  const int lane = tid & 31;
  const int wave = tid >> 5;   // 0..7
  const int wm   = wave & 1;   // wave row  (2)
  const int wn   = wave >> 1;  // wave col  (4)
  const int half = lane >> 4;
  const int l16  = lane & 15;

  const int bm = blockIdx.x * BM;
  const int bn = blockIdx.y * BN;

  const int ar = wm * 32 + l16;  // A fragment base row in LDS x-tile
  const int br = wn * 32 + l16;  // B fragment base row in LDS W-tile (out col)

  v8f acc[2][2] = {};

  // Per-thread staging coordinates (16B chunks; 8 float4 per 32-float row).
  // x tile: 64x32 = 512 float4 -> 2 per thread; W tile: 128x32 -> 4 per thread.
  auto issue_tile = [&](int buf, int k0) {
#pragma unroll
    for (int i = 0; i < 2; ++i) {
      const int flat = tid + i * 256;
      const int row = flat >> 3, c4 = flat & 7;
      async_b128(&xs[buf][row * LDS_STRIDE + c4 * 4],
                 x + (size_t)(bm + row) * IN_F + k0 + c4 * 4);
    }
#pragma unroll
    for (int i = 0; i < 4; ++i) {
      const int flat = tid + i * 256;
      const int row = flat >> 3, c4 = flat & 7;
      async_b128(&ws[buf][row * LDS_STRIDE + c4 * 4],
                 W + (size_t)(bn + row) * IN_F + k0 + c4 * 4);
    }
  };

  issue_tile(0, 0);

  const int NT = IN_F / KT;  // 128 K-tiles
  for (int it = 0; it < NT; ++it) {
    const int cur = it & 1;
    if (it + 1 < NT) {
      issue_tile(cur ^ 1, (it + 1) * KT);
      wait_async_le6();  // pend only this wave's 6 ops for the *current* tile
    } else {
      wait_async_0();
    }
    // prefetch two tiles ahead into GL2 (global_prefetch_b8)
    if (it + 2 < NT) {
      const int row = tid >> 3, c4 = tid & 7;
      __builtin_prefetch(x + (size_t)(bm + row) * IN_F + (it + 2) * KT + c4 * 4, 0, 0);
      __builtin_prefetch(W + (size_t)(bn + row) * IN_F + (it + 2) * KT + c4 * 4, 0, 0);
    }
    __syncthreads();

#pragma unroll
    for (int kk = 0; kk < KT; kk += 4) {
      const int kb = kk + half * 2;
      v2f a0 = *(const v2f*)(&xs[cur][(ar)      * LDS_STRIDE + kb]);
      v2f a1 = *(const v2f*)(&xs[cur][(ar + 16) * LDS_STRIDE + kb]);
      v2f b0 = *(const v2f*)(&ws[cur][(br)      * LDS_STRIDE + kb]);
      v2f b1 = *(const v2f*)(&ws[cur][(br + 16) * LDS_STRIDE + kb]);
      acc[0][0] = wmma_f32(a0, b0, acc[0][0]);
      acc[0][1] = wmma_f32(a0, b1, acc[0][1]);
      acc[1][0] = wmma_f32(a1, b0, acc[1][0]);
      acc[1][1] = wmma_f32(a1, b1, acc[1][1]);
    }
    __syncthreads();  // buffer `cur` free for reuse by iteration it+2
  }

  // ---- LoRA delta: one extra rank-16 K-block, operands straight from L2 ----
  {
    const float* t0 = t     + (size_t)(bm + wm * 32 + l16) * RANK;
    const float* t1 = t0    + (size_t)16 * RANK;
    const float* A0 = loraA + (size_t)(bn + wn * 32 + l16) * RANK;
    const float* A1 = A0    + (size_t)16 * RANK;
#pragma unroll
    for (int kk = 0; kk < RANK; kk += 4) {
      const int kb = kk + half * 2;
      v2f a0; a0.x = t0[kb]; a0.y = t0[kb + 1];
      v2f a1; a1.x = t1[kb]; a1.y = t1[kb + 1];
      v2f b0; b0.x = A0[kb]; b0.y = A0[kb + 1];
      v2f b1; b1.x = A1[kb]; b1.y = A1[kb + 1];
      acc[0][0] = wmma_f32(a0, b0, acc[0][0]);
      acc[0][1] = wmma_f32(a0, b1, acc[0][1]);
      acc[1][0] = wmma_f32(a1, b0, acc[1][0]);
      acc[1][1] = wmma_f32(a1, b1, acc[1][1]);
    }
  }

  // ---- bias + store (half-wave writes 64B contiguous segments) ----
  const float bias0 = bias[bn + wn * 32 + l16];
  const float bias1 = bias[bn + wn * 32 + 16 + l16];
#pragma unroll
  for (int i = 0; i < 2; ++i) {
    const int mbase = bm + wm * 32 + i * 16 + half * 8;
#pragma unroll
    for (int v = 0; v < 8; ++v) {
      float* orow = out + (size_t)(mbase + v) * OUT_F + bn + wn * 32 + l16;
      orow[0]  = acc[i][0][v] + bias0;
      orow[16] = acc[i][1][v] + bias1;
    }
  }
}

// ---------------------------------------------------------------------------
extern "C" void kernel_launch(void* const* d_in, const int* in_sizes, int n_in,
                              void* d_out, int out_size, void* d_ws, size_t ws_size,
                              hipStream_t stream) {
  (void)in_sizes; (void)n_in; (void)out_size; (void)ws_size;
  const float* x     = (const float*)d_in[0];
  const float* W     = (const float*)d_in[1];
  const float* bias  = (const float*)d_in[2];
  const float* loraA = (const float*)d_in[3];
  const float* loraB = (const float*)d_in[4];
  float* out = (float*)d_out;
  float* t   = (float*)d_ws;  // M_TOK * RANK * 4 = 512 KiB scratch

  lora_t_kernel<<<M_TOK / 16, 32, 0, stream>>>(x, loraB, t);

  dim3 grid(M_TOK / BM, OUT_F / BN);  // 128 x 32
  lora_gemm_kernel<<<grid, 256, 0, stream>>>(x, W, bias, loraA, t, out);
}